// ChamferDistanceL1_85564338471044
// MI455X (gfx1250) — compile-verified
//
#include <hip/hip_runtime.h>
#include <hip/hip_bf16.h>

// Problem constants (from reference: pred/gt are [4, 4096, 3] f32)
#define BB    4
#define NP    4096
#define TPB   256            // threads per block == query points per block
#define TG    1024           // reference points staged in LDS per chunk
#define SPLIT 2              // reference-dim split across blocks
#define CPB   (NP / SPLIT / TG)  // LDS chunks per block (2)

#if __has_builtin(__builtin_amdgcn_tensor_load_to_lds)
#define HAVE_TDM 1
#else
#define HAVE_TDM 0
#endif

typedef unsigned int u32x4 __attribute__((ext_vector_type(4)));
typedef int          i32x4 __attribute__((ext_vector_type(4)));
typedef int          i32x8 __attribute__((ext_vector_type(8)));

#if HAVE_TDM
// One TDM DMA: copy `ndwords` DWORDs from global `gsrc` into LDS at byte
// offset `lds_off`, expressed as a 1-row 2D tile in the D#.
// Tracked by TENSORcnt; caller must s_wait_tensorcnt + barrier before use.
__device__ __forceinline__ void tdm_load_chunk(unsigned int lds_off,
                                               const float* gsrc,
                                               int ndwords) {
  unsigned long long ga = (unsigned long long)(uintptr_t)gsrc;

  // D# group 0: [1:0]=count=1, [63:32]=lds_addr, [120:64]=global_addr,
  // [127:126]=type=2 ("image")
  u32x4 g0;
  g0.x = 1u;
  g0.y = lds_off;
  g0.z = (unsigned int)ga;
  g0.w = (unsigned int)((ga >> 32) & 0x01FFFFFFull) | (2u << 30);

  // D# group 1: wg_mask=0, data_size=2 (4B), tensor_dim0=tile_dim0=ndwords,
  // tensor_dim1=tile_dim1=1, strides=ndwords (1-row tile), no pad.
  unsigned int nd = (unsigned int)ndwords;
  i32x8 g1;
  g1[0] = (int)0x00020000u;                            // data_size=4B
  g1[1] = (int)((nd & 0xFFFFu) << 16);                 // tensor_dim0 lo16
  g1[2] = (int)(((nd >> 16) & 0xFFFFu) | (1u << 16));  // dim0 hi | tensor_dim1=1
  g1[3] = (int)((nd & 0xFFFFu) << 16);                 // dim1 hi=0 | tile_dim0
  g1[4] = 1;                                           // tile_dim1=1, tile_dim2=0
  g1[5] = (int)nd;                                     // tensor_dim0_stride lo32
  g1[6] = (int)((nd & 0xFFFFu) << 16);                 // stride0 hi=0 | stride1 lo16
  g1[7] = 0;                                           // stride1 hi

  i32x4 gz = {0, 0, 0, 0};
#if defined(__clang_major__) && (__clang_major__ >= 23)
  i32x8 gz8 = {0, 0, 0, 0, 0, 0, 0, 0};
  __builtin_amdgcn_tensor_load_to_lds(g0, g1, gz, gz, gz8, 0);
#else
  __builtin_amdgcn_tensor_load_to_lds(g0, g1, gz, gz, 0);
#endif
}
#endif // HAVE_TDM

// Initialize per-point minima to +inf bit pattern (for atomic u32 min).
__global__ __launch_bounds__(TPB) void chamfer_init_kernel(
    unsigned int* __restrict__ ws_min_u) {
  int i = blockIdx.x * TPB + threadIdx.x;
  if (i < 2 * BB * NP) ws_min_u[i] = 0x7F800000u;  // +inf
}

// One block: 256 query points (blockIdx.x), batch b (blockIdx.y),
// {direction, reference-half} (blockIdx.z). Loops over its 2048-point
// reference slice in LDS chunks of 1024, double-buffered via the Tensor
// Data Mover. Per-point running min merged globally with atomic u32 min
// (valid: distances >= 0, so f32 ordering == u32 ordering of the bits).
__global__ __launch_bounds__(TPB) void chamfer_min_kernel(
    const float* __restrict__ pred, const float* __restrict__ gt,
    unsigned int* __restrict__ ws_min_u) {
  __shared__ float s_g[2][TG * 3];

  const int dir  = blockIdx.z / SPLIT;  // 0: min over gt per pred; 1: swapped
  const int half = blockIdx.z % SPLIT;  // which half of the reference set
  const int b    = blockIdx.y;
  const int tid  = threadIdx.x;
  const int n    = blockIdx.x * TPB + tid;

  const float* __restrict__ P = dir ? gt : pred;
  const float* __restrict__ G = dir ? pred : gt;

  const float* __restrict__ gbase =
      G + ((size_t)b * NP + (size_t)half * (NP / SPLIT)) * 3;

  // Prime buffer 0 with chunk 0 ASAP so the DMA overlaps the query loads.
#if HAVE_TDM
  if (tid == 0)
    tdm_load_chunk((unsigned int)(uintptr_t)&s_g[0][0], gbase, TG * 3);
#else
  for (int i = tid; i < TG * 3; i += TPB) s_g[0][i] = gbase[i];
#endif

  const float px = P[((size_t)b * NP + n) * 3 + 0];
  const float py = P[((size_t)b * NP + n) * 3 + 1];
  const float pz = P[((size_t)b * NP + n) * 3 + 2];

#if HAVE_TDM
  if (tid == 0) __builtin_amdgcn_s_wait_tensorcnt(0);
#endif
  __syncthreads();

  float mn = __FLT_MAX__;
  for (int ch = 0; ch < CPB; ++ch) {
    const int cur = ch & 1;
    const int nxt = cur ^ 1;

#if HAVE_TDM
    // Kick off DMA of the next chunk while consuming the current one.
    // TDM ignores EXEC; uniform branch means only wave 0 reaches it, and it
    // issues exactly once per chunk.
    if ((ch + 1 < CPB) && tid == 0)
      tdm_load_chunk((unsigned int)(uintptr_t)&s_g[nxt][0],
                     gbase + (size_t)(ch + 1) * TG * 3, TG * 3);
#endif

    const float* sg = &s_g[cur][0];
#pragma unroll 4
    for (int j = 0; j < TG; ++j) {
      // All 32 lanes read the same LDS address -> broadcast, conflict-free.
      float dx = px - sg[3 * j + 0];
      float dy = py - sg[3 * j + 1];
      float dz = pz - sg[3 * j + 2];
      float d  = __builtin_fabsf(dx) + __builtin_fabsf(dy) +
                 __builtin_fabsf(dz);
      mn = fminf(mn, d);
    }

    if (ch + 1 < CPB) {
#if HAVE_TDM
      if (tid == 0) __builtin_amdgcn_s_wait_tensorcnt(0);
      __syncthreads();  // publish next buffer to all waves
#else
      __syncthreads();  // all reads of s_g done
      for (int i = tid; i < TG * 3; i += TPB)
        s_g[nxt][i] = gbase[(size_t)(ch + 1) * TG * 3 + i];
      __syncthreads();
#endif
    }
  }

  // Merge this block's partial min (distances >= 0 -> u32-order == f32-order).
  atomicMin(&ws_min_u[((size_t)dir * BB + b) * NP + n], __float_as_uint(mn));
}

// Since N == M, mean_b(cham_x + cham_y) == sum(all 2*B*N minima) / (B*N).
__global__ __launch_bounds__(256) void chamfer_reduce_kernel(
    const float* __restrict__ ws_min, float* __restrict__ out) {
  __shared__ float red[256];
  float s = 0.0f;
  for (int i = threadIdx.x; i < 2 * BB * NP; i += 256) s += ws_min[i];
  red[threadIdx.x] = s;
  __syncthreads();
  for (int off = 128; off > 0; off >>= 1) {
    if ((int)threadIdx.x < off) red[threadIdx.x] += red[threadIdx.x + off];
    __syncthreads();
  }
  if (threadIdx.x == 0) out[0] = red[0] / (float)(BB * NP);
}

extern "C" void kernel_launch(void* const* d_in, const int* in_sizes, int n_in,
                              void* d_out, int out_size, void* d_ws,
                              size_t ws_size, hipStream_t stream) {
  const float* pred = (const float*)d_in[0];  // [4,4096,3] f32
  const float* gt   = (const float*)d_in[1];  // [4,4096,3] f32
  unsigned int* ws_u = (unsigned int*)d_ws;   // 2*B*N u32/f32 = 128 KB
  (void)in_sizes; (void)n_in; (void)out_size; (void)ws_size;

  const int nmin = 2 * BB * NP;
  chamfer_init_kernel<<<(nmin + TPB - 1) / TPB, TPB, 0, stream>>>(ws_u);

  dim3 grid(NP / TPB, BB, 2 * SPLIT);
  chamfer_min_kernel<<<grid, TPB, 0, stream>>>(pred, gt, ws_u);

  chamfer_reduce_kernel<<<1, 256, 0, stream>>>((const float*)d_ws,
                                               (float*)d_out);
}